// LoraLayer_58050777973155
// MI455X (gfx1250) — compile-verified
//
#include <hip/hip_runtime.h>

typedef float v2f __attribute__((ext_vector_type(2)));
typedef float v8f __attribute__((ext_vector_type(8)));

#define BS    4096
#define H     2048
#define OUTD  2048
#define MMOD  3
#define SSLOT 4
#define RMAX  64
#define KC    64            // K chunk (GEMM1) / col chunk (GEMM2)
#define LDP   68            // padded LDS row stride in dwords (== 4 mod 64 -> conflict-free b64 frags)
#define ROWS  32            // token rows per block

__global__ __launch_bounds__(256)
void lora_fused_wmma(const float* __restrict__ x,
                     const float* __restrict__ lora_A,
                     const float* __restrict__ lora_B,
                     const int*   __restrict__ sorted_ids,
                     const int*   __restrict__ row_slot,
                     const int*   __restrict__ slot_ranks,
                     float*       __restrict__ out)
{
    __shared__ float Xlds[ROWS * LDP];   // gathered X tile, [row][k]
    __shared__ float Blds[RMAX * LDP];   // GEMM1: loraA chunk [r][k]; GEMM2: loraB chunk transposed [c][r]
    __shared__ float Ilds[ROWS * LDP];   // intermediate [row][r]
    __shared__ int   tok[ROWS];

    const int tid  = threadIdx.x;
    const int lane = tid & 31;
    const int wave = tid >> 5;
    const int rowbase = blockIdx.x * ROWS;
    const int m = blockIdx.y;
    const int outStride = MMOD * OUTD;   // 6144

    if (tid < ROWS) tok[tid] = sorted_ids[rowbase + tid];
    __syncthreads();

    const int slot = row_slot[rowbase];  // uniform within a 32-row tile (boundaries are multiples of 32)

    if (slot < 0) {
        // tail rows: no adapter -> this module's slice of the output is zero.
        // Non-temporal: write-once output, keep L2 for reusable operands.
        for (int i = tid; i < ROWS * OUTD; i += 256) {
            int lr = i >> 11;            // / 2048
            int c  = i & (OUTD - 1);
            __builtin_nontemporal_store(0.0f,
                &out[(size_t)tok[lr] * outStride + m * OUTD + c]);
        }
        return;
    }

    const int rank      = slot_ranks[slot];   // in {16,32,48,64}
    const int rankTiles = rank >> 4;

    const float* gA = lora_A + (size_t)(m * SSLOT + slot) * RMAX * H;
    const float* gB = lora_B + (size_t)(m * SSLOT + slot) * RMAX * OUTD;

    const int rowTile = wave >> 2;       // 0..1 : which 16-row half
    const int subTile = wave & 3;        // 0..3 : rank tile (GEMM1) / col subtile (GEMM2)
    const int halfsel = lane >> 4;       // 0/1
    const int l16     = lane & 15;
    const int kdelta  = 2 * halfsel;     // A/B frag K offset per ISA 16x16x4 f32 layout

    // ---------------- GEMM1: inter[32 x rank] = Xtile @ loraA^T ----------------
    v8f acc = {};
    const bool rtActive = (subTile < rankTiles);   // wave-uniform

    for (int kb = 0; kb < H; kb += KC) {
        // stage gathered X rows (32 x 64 fp32) as float4
        for (int i = tid; i < ROWS * (KC / 4); i += 256) {
            int lr = i >> 4, q = i & 15;
            float4 v = *(const float4*)(x + (size_t)tok[lr] * H + kb + 4 * q);
            *(float4*)(&Xlds[lr * LDP + 4 * q]) = v;
        }
        // stage loraA chunk [rank x 64], memory layout is already k-contiguous
        for (int i = tid; i < rank * (KC / 4); i += 256) {
            int r = i >> 4, q = i & 15;
            float4 v = *(const float4*)(gA + (size_t)r * H + kb + 4 * q);
            *(float4*)(&Blds[r * LDP + 4 * q]) = v;
        }
        __syncthreads();

        if (rtActive) {
            const float* xrow = &Xlds[(rowTile * 16 + l16) * LDP];
            const float* brow = &Blds[(subTile * 16 + l16) * LDP];
#pragma unroll
            for (int k0 = 0; k0 < KC; k0 += 4) {
                v2f a = *(const v2f*)(xrow + k0 + kdelta);
                v2f b = *(const v2f*)(brow + k0 + kdelta);
                acc = __builtin_amdgcn_wmma_f32_16x16x4_f32(
                        false, a, false, b, (short)0, acc, false, false);
            }
        }
        __syncthreads();
    }

    // spill intermediate to LDS (columns >= rank are never read in GEMM2)
    if (rtActive) {
#pragma unroll
        for (int j = 0; j < 8; ++j) {
            int r = rowTile * 16 + j + 8 * halfsel;
            Ilds[r * LDP + subTile * 16 + l16] = acc[j];
        }
    }
    __syncthreads();

    // ---------------- GEMM2: out[32 x 2048] = inter @ loraB ----------------
    for (int cb = 0; cb < OUTD; cb += KC) {
        // stage loraB chunk transposed: Blds[c][r] = gB[r][cb+c]  (coalesced global reads)
        for (int i = tid; i < rank * KC; i += 256) {
            int r = i >> 6, c = i & 63;
            Blds[c * LDP + r] = gB[(size_t)r * OUTD + cb + c];
        }
        __syncthreads();

        v8f acc2 = {};
        const float* irow  = &Ilds[(rowTile * 16 + l16) * LDP];
        const float* b2row = &Blds[(subTile * 16 + l16) * LDP];
        for (int k0 = 0; k0 < rank; k0 += 4) {   // wave-uniform trip count
            v2f a = *(const v2f*)(irow + k0 + kdelta);
            v2f b = *(const v2f*)(b2row + k0 + kdelta);
            acc2 = __builtin_amdgcn_wmma_f32_16x16x4_f32(
                    false, a, false, b, (short)0, acc2, false, false);
        }

        // scatter 16x16 tile back to token order; non-temporal (write-once, 100MB stream)
#pragma unroll
        for (int j = 0; j < 8; ++j) {
            int lr = rowTile * 16 + j + 8 * halfsel;
            __builtin_nontemporal_store(acc2[j],
                &out[(size_t)tok[lr] * outStride + m * OUTD + cb + subTile * 16 + l16]);
        }
        __syncthreads();
    }
}

extern "C" void kernel_launch(void* const* d_in, const int* in_sizes, int n_in,
                              void* d_out, int out_size, void* d_ws, size_t ws_size,
                              hipStream_t stream)
{
    const float* x          = (const float*)d_in[0];
    const float* lora_A     = (const float*)d_in[1];
    const float* lora_B     = (const float*)d_in[2];
    const int*   sorted_ids = (const int*)d_in[3];
    const int*   row_slot   = (const int*)d_in[4];
    const int*   slot_ranks = (const int*)d_in[5];
    float*       out        = (float*)d_out;

    dim3 grid(BS / ROWS, MMOD);   // 128 x 3 = 384 blocks, 256 threads (8 wave32) each
    lora_fused_wmma<<<grid, 256, 0, stream>>>(x, lora_A, lora_B,
                                              sorted_ids, row_slot, slot_ranks, out);
}